// ASPP_73418170958021
// MI455X (gfx1250) — compile-verified
//
#include <hip/hip_runtime.h>

// Problem constants (from reference)
#define NPTS   100000
#define NB     4
#define NC     32
#define NK     27
#define NM     30000
#define NWIDTH 5
#define CATC   ((NWIDTH + 2) * NC)   // 224
#define EPSB   1e-5f

// Use CDNA5 async global->LDS DMA path (ASYNCcnt) for weight staging.
#define USE_ASYNC_STAGE 1

typedef __attribute__((ext_vector_type(2))) float v2f;
typedef __attribute__((ext_vector_type(4))) float v4f;
typedef __attribute__((ext_vector_type(8))) float v8f;

// ---------------------------------------------------------------------------
// Weight layout: row-pair interleaved so each WMMA B-operand is one contiguous
// 8-byte ds_load_b64:
//   Wi[(r>>1)*64 + c*2 + (r&1)] = W[r][c]      (W row-major [CIN][32])
// k_prep writes interleaved copies of all weights into the workspace once per
// call; staging to LDS is then a contiguous bulk copy (async-to-LDS capable).
// ---------------------------------------------------------------------------

// Contiguous global->LDS copy of nfloats (multiple of 4), block-cooperative.
__device__ __forceinline__ void stage_copy(float* __restrict__ dstLds,
                                           const float* __restrict__ srcGlb,
                                           int nfloats) {
#if USE_ASYNC_STAGE
    for (int c = threadIdx.x * 4; c < nfloats; c += blockDim.x * 4) {
        const unsigned lds = (unsigned)(unsigned long long)(dstLds + c);
        const float* g = srcGlb + c;
        asm volatile("global_load_async_to_lds_b128 %0, %1, off"
                     :: "v"(lds), "v"(g) : "memory");
    }
    asm volatile("s_wait_asynccnt 0x0" ::: "memory");
#else
    for (int c = threadIdx.x * 4; c < nfloats; c += blockDim.x * 4)
        *(v4f*)(dstLds + c) = *(const v4f*)(srcGlb + c);
#endif
    // caller must __syncthreads()
}

// ---------------------------------------------------------------------------
// Wave-level 16-edge x Cin x 32 tile GEMM using V_WMMA_F32_16X16X4_F32.
// VGPR layouts per CDNA5 ISA 7.12.2:
//   A 16x4 : lane 16h+s holds A[s][2h], A[s][2h+1] in its 2 VGPRs
//   B  4x16: lane 16h+s holds B[2h][s], B[2h+1][s]
//   D 16x16: lane 16h+s, VGPR r holds D[8h+r][s]
// EXEC must be all-1s: callers only guard with wave-uniform conditions.
// ---------------------------------------------------------------------------
template <int CIN, bool ATOMIC, bool IDENT>
__device__ __forceinline__ void wave_tile(
    const float* __restrict__ X, int ldX,
    const float* __restrict__ Wp,           // LDS, interleaved
    const int* __restrict__ in_map,         // 16 gather indices (ignored if IDENT)
    const int* __restrict__ out_map,        // 16 scatter indices (ignored if IDENT)
    int identBase,
    float* __restrict__ Y, int ldY,
    const float* __restrict__ bias)         // [32] or nullptr (only if !ATOMIC)
{
    const int lane = threadIdx.x & 31;
    const int half = lane >> 4;             // 0 or 1
    const int sub  = lane & 15;

    const int row = IDENT ? (identBase + sub) : in_map[sub];
    const float* xr = X + (long)row * ldX + half * 2;  // 8B aligned (ldX even)
    const float* wl = Wp + half * 64 + sub * 2;        // lane's B base in LDS

    v8f acc0 = {0.f, 0.f, 0.f, 0.f, 0.f, 0.f, 0.f, 0.f};
    v8f acc1 = {0.f, 0.f, 0.f, 0.f, 0.f, 0.f, 0.f, 0.f};

#pragma unroll
    for (int kk = 0; kk < CIN / 4; ++kk) {
        const v2f a  = *(const v2f*)(xr + kk * 4);         // global_load_b64
        const v2f b0 = *(const v2f*)(wl + kk * 128);       // ds_load_b64 (cols 0..15)
        const v2f b1 = *(const v2f*)(wl + kk * 128 + 32);  // ds_load_b64 (cols 16..31)
        acc0 = __builtin_amdgcn_wmma_f32_16x16x4_f32(
            false, a, false, b0, (short)0, acc0, false, false);
        acc1 = __builtin_amdgcn_wmma_f32_16x16x4_f32(
            false, a, false, b1, (short)0, acc1, false, false);
    }

    float bias0 = 0.f, bias1 = 0.f;
    if (!ATOMIC && bias) { bias0 = bias[sub]; bias1 = bias[16 + sub]; }

#pragma unroll
    for (int r = 0; r < 8; ++r) {
        const int orow = IDENT ? (identBase + 8 * half + r) : out_map[8 * half + r];
        float* yp = Y + (long)orow * ldY;
        if (ATOMIC) {
            atomicAdd(yp + sub,      acc0[r]);
            atomicAdd(yp + 16 + sub, acc1[r]);
        } else {
            yp[sub]      = acc0[r] + bias0;
            yp[16 + sub] = acc1[r] + bias1;
        }
    }
}

// ---------------------------------------------------------------------------
// Kernels
// ---------------------------------------------------------------------------
__global__ void k_init(float* __restrict__ p, long n) {
    long i = (long)blockIdx.x * blockDim.x + threadIdx.x;
    const long stride = (long)gridDim.x * blockDim.x;
    for (; i < n; i += stride) p[i] = 0.f;
}

// Write row-pair-interleaved copies of W_lin, W_branch, W_out into workspace.
__global__ void k_prep(const float* __restrict__ Wlin, const float* __restrict__ Wb,
                       const float* __restrict__ Wout, float* __restrict__ Wi) {
    const long nLin = (long)NC * NC;
    const long nBr  = (long)NWIDTH * NK * NC * NC;
    const long nOut = (long)NK * CATC * NC;
    const long idx  = (long)blockIdx.x * blockDim.x + threadIdx.x;
    if (idx < nLin) {
        const int r = (int)(idx >> 5), c = (int)(idx & 31);
        Wi[(r >> 1) * 64 + c * 2 + (r & 1)] = Wlin[idx];
    } else if (idx < nLin + nBr) {
        const long t = idx - nLin;
        const long mat = t / (NC * NC);
        const int  e = (int)(t % (NC * NC));
        const int  r = e >> 5, c = e & 31;
        Wi[nLin + mat * (NC * NC) + (r >> 1) * 64 + c * 2 + (r & 1)] = Wb[t];
    } else if (idx < nLin + nBr + nOut) {
        const long t = idx - nLin - nBr;
        const long mat = t / (CATC * NC);
        const int  e = (int)(t % (CATC * NC));
        const int  r = e >> 5, c = e & 31;
        Wi[nLin + nBr + mat * (CATC * NC) + (r >> 1) * 64 + c * 2 + (r & 1)] = Wout[t];
    }
}

// Per-batch segment sum of x + counts, block-local LDS reduction then global atomics.
__global__ void k_pool(const float* __restrict__ x, const int* __restrict__ bidx,
                       float* __restrict__ sums, float* __restrict__ cnts) {
    __shared__ float s_sum[NB * NC];
    __shared__ float s_cnt[NB];
    const int t = threadIdx.x;
    if (t < NB * NC) s_sum[t] = 0.f;
    if (t < NB)      s_cnt[t] = 0.f;
    __syncthreads();
    const int i = blockIdx.x * blockDim.x + t;
    if (i < NPTS) {
        const int b = bidx[i];
        const float* xr = x + (long)i * NC;
        atomicAdd(&s_cnt[b], 1.f);
        for (int c = 0; c < NC; ++c) atomicAdd(&s_sum[b * NC + c], xr[c]);
    }
    __syncthreads();
    if (t < NB * NC) atomicAdd(&sums[t], s_sum[t]);
    if (t < NB)      atomicAdd(&cnts[t], s_cnt[t]);
}

// cat[:, 0:32] = x @ W_lin + b_lin  (identity maps, direct store)
__global__ void k_linear(const float* __restrict__ x, const float* __restrict__ WiLin,
                         const float* __restrict__ blin, float* __restrict__ cat) {
    __shared__ alignas(16) float Wp[NC * NC];
    stage_copy(Wp, WiLin, NC * NC);
    __syncthreads();
    const int wave   = threadIdx.x >> 5;
    const int stride = gridDim.x * (blockDim.x >> 5);
    for (int tile = blockIdx.x * (blockDim.x >> 5) + wave; tile < NPTS / 16; tile += stride) {
        wave_tile<NC, /*ATOMIC=*/false, /*IDENT=*/true>(
            x, NC, Wp, nullptr, nullptr, tile * 16, cat, CATC, blin);
    }
}

// cat[:, 192:224] = (sums/counts)[batch_idx]
__global__ void k_broadcast(const int* __restrict__ bidx, const float* __restrict__ sums,
                            const float* __restrict__ cnts, float* __restrict__ cat) {
    const long idx = (long)blockIdx.x * blockDim.x + threadIdx.x;
    if (idx < (long)NPTS * NC) {
        const int i = (int)(idx >> 5);
        const int c = (int)(idx & 31);
        const int b = bidx[i];
        cat[(long)i * CATC + (NWIDTH + 1) * NC + c] = sums[b * NC + c] / cnts[b];
    }
}

// 5 dilated branch convs: gather x, 32x32 GEMM per offset, atomic scatter into cat.
// blockIdx.y = branch*NK + k ; W (4 KB, pre-interleaved) staged async into LDS.
__global__ void k_branch(const float* __restrict__ x, const float* __restrict__ WiBr,
                         const int* __restrict__ in_maps, const int* __restrict__ out_maps,
                         float* __restrict__ cat) {
    __shared__ alignas(16) float Wp[NC * NC];
    const int bk = blockIdx.y;              // branch*NK + k
    const int branch = bk / NK;
    stage_copy(Wp, WiBr + (long)bk * NC * NC, NC * NC);
    __syncthreads();
    const int wave   = threadIdx.x >> 5;
    const int stride = gridDim.x * (blockDim.x >> 5);
    for (int tile = blockIdx.x * (blockDim.x >> 5) + wave; tile < NM / 16; tile += stride) {
        const long mbase = (long)bk * NM + (long)tile * 16;
        wave_tile<NC, /*ATOMIC=*/true, /*IDENT=*/false>(
            x, NC, Wp, in_maps + mbase, out_maps + mbase, 0,
            cat + (long)(branch + 1) * NC, CATC, nullptr);
    }
}

// Output conv: gather cat (224 ch), 224x32 GEMM per offset, atomic scatter into y.
// blockIdx.y = k ; W_out[k] (28 KB, pre-interleaved) staged async into LDS.
__global__ void k_out(const float* __restrict__ cat, const float* __restrict__ WiOut,
                      const int* __restrict__ in_maps, const int* __restrict__ out_maps,
                      float* __restrict__ y) {
    __shared__ alignas(16) float Wp[CATC * NC];
    const int k = blockIdx.y;
    stage_copy(Wp, WiOut + (long)k * CATC * NC, CATC * NC);
    __syncthreads();
    const int wave   = threadIdx.x >> 5;
    const int stride = gridDim.x * (blockDim.x >> 5);
    for (int tile = blockIdx.x * (blockDim.x >> 5) + wave; tile < NM / 16; tile += stride) {
        const long mbase = ((long)NWIDTH * NK + k) * NM + (long)tile * 16;
        wave_tile<CATC, /*ATOMIC=*/true, /*IDENT=*/false>(
            cat, CATC, Wp, in_maps + mbase, out_maps + mbase, 0, y, NC, nullptr);
    }
}

__global__ void k_bnrelu(const float* __restrict__ y,
                         const float* __restrict__ gamma, const float* __restrict__ beta,
                         const float* __restrict__ mean,  const float* __restrict__ var,
                         float* __restrict__ out) {
    const long idx = (long)blockIdx.x * blockDim.x + threadIdx.x;
    if (idx < (long)NPTS * NC) {
        const int c = (int)(idx & 31);
        const float v = (y[idx] - mean[c]) * rsqrtf(var[c] + EPSB) * gamma[c] + beta[c];
        out[idx] = v > 0.f ? v : 0.f;
    }
}

// ---------------------------------------------------------------------------
extern "C" void kernel_launch(void* const* d_in, const int* in_sizes, int n_in,
                              void* d_out, int out_size, void* d_ws, size_t ws_size,
                              hipStream_t stream) {
    const float* x     = (const float*)d_in[0];
    const int*   bidx  = (const int*)  d_in[1];
    const int*   inm   = (const int*)  d_in[2];
    const int*   outm  = (const int*)  d_in[3];
    const float* Wlin  = (const float*)d_in[4];
    const float* blin  = (const float*)d_in[5];
    const float* Wb    = (const float*)d_in[6];
    const float* Wout  = (const float*)d_in[7];
    const float* gamma = (const float*)d_in[8];
    const float* beta  = (const float*)d_in[9];
    const float* bmean = (const float*)d_in[10];
    const float* bvar  = (const float*)d_in[11];

    // Workspace (fp32): cat [N][224] | y [N][32] | sums [B][32] | cnts [B] | Wi
    float* cat  = (float*)d_ws;
    float* ybuf = cat  + (long)NPTS * CATC;
    float* sums = ybuf + (long)NPTS * NC;
    float* cnts = sums + NB * NC;
    float* Wi   = cnts + NB;                 // 16B-aligned (offset = 33*16 B mod 16)
    const long zn = (long)NPTS * CATC + (long)NPTS * NC + NB * NC + NB;

    const long nLin = (long)NC * NC;                      // 1024
    const long nBr  = (long)NWIDTH * NK * NC * NC;        // 138240
    const long nOut = (long)NK * CATC * NC;               // 193536
    const long nW   = nLin + nBr + nOut;                  // 332800

    k_init<<<2048, 256, 0, stream>>>((float*)d_ws, zn);
    k_prep<<<(int)((nW + 255) / 256), 256, 0, stream>>>(Wlin, Wb, Wout, Wi);
    k_pool<<<(NPTS + 255) / 256, 256, 0, stream>>>(x, bidx, sums, cnts);
    k_linear<<<128, 256, 0, stream>>>(x, Wi, blin, cat);
    k_broadcast<<<(int)(((long)NPTS * NC + 255) / 256), 256, 0, stream>>>(bidx, sums, cnts, cat);

    dim3 gb(64, NWIDTH * NK);                // 64 x 135, ~3.7 tiles/wave
    k_branch<<<gb, 256, 0, stream>>>(x, Wi + nLin, inm, outm, cat);

    dim3 go(64, NK);                         // 64 x 27
    k_out<<<go, 256, 0, stream>>>(cat, Wi + nLin + nBr, inm, outm, ybuf);

    k_bnrelu<<<(int)(((long)NPTS * NC + 255) / 256), 256, 0, stream>>>(
        ybuf, gamma, beta, bmean, bvar, (float*)d_out);
}